// EQCNN_equi_unet_65154653880709
// MI455X (gfx1250) — compile-verified
//
#include <hip/hip_runtime.h>
#include <math.h>

#define EPSc    1e-6f
#define BN_EPSc 1e-5f
#define NSc     0.2f

typedef float v2f __attribute__((ext_vector_type(2)));
typedef float v8f __attribute__((ext_vector_type(8)));

#if defined(__HIP_DEVICE_COMPILE__)
#if !__has_builtin(__builtin_amdgcn_wmma_f32_16x16x4_f32)
#error "wmma f32 16x16x4 builtin missing on this toolchain"
#endif
#endif

#define WMMA4(A, B, C) __builtin_amdgcn_wmma_f32_16x16x4_f32(false, A, false, B, (short)0, C, false, false)

// ---------------------------------------------------------------------------
// CDNA5 async global->LDS copy (ASYNCcnt-tracked), CDNA5 ISA.
// Each active lane copies 4 bytes from global memory into LDS.
// ---------------------------------------------------------------------------
__device__ __forceinline__ void async_g2l_b32(float* lds_dst, const float* gsrc) {
  unsigned loff = (unsigned)(uintptr_t)lds_dst;   // low 32 bits = LDS offset
  asm volatile("global_load_async_to_lds_b32 %0, %1, off"
               :: "v"(loff), "v"(gsrc) : "memory");
}
__device__ __forceinline__ void wait_async0() {
  asm volatile("s_wait_asynccnt 0x0" ::: "memory");
}

// ---------------------------------------------------------------------------
// General (runtime) B fetch -- used only in the ragged K tail (conv1, Cin=2).
// X is (Csrc, 3, Nsrc) row-major.
// ---------------------------------------------------------------------------
template <int MODE>
__device__ __forceinline__ float fetchB(const float* __restrict__ X,
                                        int Csrc, int Nsrc, int c2, int d,
                                        int n, int j, int s) {
  if (MODE == 1) {
    if (c2 < Csrc) {
      const float* row = X + (c2 * 3 + d) * Nsrc;
      return row[j] - row[n];
    }
    return X[((c2 - Csrc) * 3 + d) * Nsrc + n];
  } else if (MODE == 2) {
    return X[(c2 * 3 + d) * Nsrc + j];
  }
  return X[(c2 * 3 + d) * Nsrc + s];
}

// ---------------------------------------------------------------------------
// Shared WMMA K-loop: 3 (or 6) f32 16x16x4 accumulator chains, branch-free.
// Arow points into the LDS-staged weight band; B operands are gathered from
// global (graph/group fusion) inside the loop.
// ---------------------------------------------------------------------------
template <int MODE, bool WITHD>
__device__ __forceinline__ void vn_mm(const float* __restrict__ X,
                                      int Csrc, int Nsrc, int Cin,
                                      int n, int j, int s, int o /*hi?2:0*/,
                                      const float* Arow,
                                      const float* __restrict__ Drow, float dmask,
                                      v8f& p0, v8f& p1, v8f& p2,
                                      v8f& d0, v8f& d1, v8f& d2) {
  const int Cin4 = Cin & ~3;
  int kb = 0;

  auto step = [&](int ka, float bx0, float by0, float bx1, float by1,
                  float bx2, float by2) {
    v2f av = *(const v2f*)(Arow + ka);
    v2f b0; b0.x = bx0; b0.y = by0;
    v2f b1; b1.x = bx1; b1.y = by1;
    v2f b2; b2.x = bx2; b2.y = by2;
    p0 = WMMA4(av, b0, p0);
    p1 = WMMA4(av, b1, p1);
    p2 = WMMA4(av, b2, p2);
    if (WITHD) {
      v2f dv = *(const v2f*)(Drow + ka);
      dv.x *= dmask; dv.y *= dmask;
      d0 = WMMA4(dv, b0, d0);
      d1 = WMMA4(dv, b1, d1);
      d2 = WMMA4(dv, b2, d2);
    }
  };

  if (MODE == 1) {
    // first half: edge features (nbr - ctr).  Csrc is a multiple of 4 here
    // whenever Cin4 > 0, so the boundary never straddles a k-step.
    const int h = (Csrc <= Cin4) ? Csrc : Cin4;
    for (; kb < h; kb += 4) {
      const int ka = kb + o;
      const float* r0 = X + (ka * 3 + 0) * Nsrc;
      const float* r1 = X + (ka * 3 + 1) * Nsrc;
      const float* r2 = X + (ka * 3 + 2) * Nsrc;
      const float* q0 = r0 + 3 * Nsrc;
      const float* q1 = r1 + 3 * Nsrc;
      const float* q2 = r2 + 3 * Nsrc;
      step(ka, r0[j] - r0[n], q0[j] - q0[n],
               r1[j] - r1[n], q1[j] - q1[n],
               r2[j] - r2[n], q2[j] - q2[n]);
    }
    // second half: center features.
    for (; kb < Cin4; kb += 4) {
      const int ka = kb + o;
      const int cc = ka - Csrc;
      const float* r0 = X + (cc * 3 + 0) * Nsrc;
      const float* r1 = X + (cc * 3 + 1) * Nsrc;
      const float* r2 = X + (cc * 3 + 2) * Nsrc;
      step(ka, r0[n], r0[3 * Nsrc + n],
               r1[n], r1[3 * Nsrc + n],
               r2[n], r2[3 * Nsrc + n]);
    }
  } else {
    const int col = (MODE == 2) ? j : s;
    for (; kb < Cin4; kb += 4) {
      const int ka = kb + o;
      const float* r0 = X + (ka * 3 + 0) * Nsrc;
      const float* r1 = X + (ka * 3 + 1) * Nsrc;
      const float* r2 = X + (ka * 3 + 2) * Nsrc;
      step(ka, r0[col], r0[3 * Nsrc + col],
               r1[col], r1[3 * Nsrc + col],
               r2[col], r2[3 * Nsrc + col]);
    }
  }

  // ragged tail (only conv1: Cin == 2).  Branch-free via clamp + 0/1 mask.
  if (kb < Cin) {
    const int ka = kb + o, ka1 = ka + 1;
    const float m0 = (ka  < Cin) ? 1.f : 0.f;
    const float m1 = (ka1 < Cin) ? 1.f : 0.f;
    const int c0 = (ka  < Cin) ? ka  : 0;
    const int c1 = (ka1 < Cin) ? ka1 : 0;
    v2f av; av.x = Arow[c0] * m0; av.y = Arow[c1] * m1;
    v2f b0, b1, b2;
    b0.x = fetchB<MODE>(X, Csrc, Nsrc, c0, 0, n, j, s) * m0;
    b0.y = fetchB<MODE>(X, Csrc, Nsrc, c1, 0, n, j, s) * m1;
    b1.x = fetchB<MODE>(X, Csrc, Nsrc, c0, 1, n, j, s) * m0;
    b1.y = fetchB<MODE>(X, Csrc, Nsrc, c1, 1, n, j, s) * m1;
    b2.x = fetchB<MODE>(X, Csrc, Nsrc, c0, 2, n, j, s) * m0;
    b2.y = fetchB<MODE>(X, Csrc, Nsrc, c1, 2, n, j, s) * m1;
    p0 = WMMA4(av, b0, p0);
    p1 = WMMA4(av, b1, p1);
    p2 = WMMA4(av, b2, p2);
    if (WITHD) {
      v2f dv; dv.x = Drow[c0] * m0 * dmask; dv.y = Drow[c1] * m1 * dmask;
      d0 = WMMA4(dv, b0, d0);
      d1 = WMMA4(dv, b1, d1);
      d2 = WMMA4(dv, b2, d2);
    }
  }
}

// Cooperative async stage of the block's 16-row weight band into LDS.
__device__ __forceinline__ void stage_wband(float* sW, const float* __restrict__ W,
                                            int tileM, int Cin) {
  const float* gW = W + tileM * Cin;
  const int tot = 16 * Cin;
  for (int t = threadIdx.x; t < tot; t += 256)
    async_g2l_b32(&sW[t], gW + t);
  wait_async0();
  __syncthreads();
}

// ---------------------------------------------------------------------------
// Pass 1: fused GEMM + per-channel norm statistics (deterministic partials).
// ---------------------------------------------------------------------------
template <int MODE>
__global__ void __launch_bounds__(256)
vn_stats_kernel(const float* __restrict__ X, const int* __restrict__ idx,
                int Csrc, int Nsrc, int Cin, int N, int Kk,
                const float* __restrict__ W, int Cout,
                float* __restrict__ partial, int WTOT) {
  extern __shared__ float sW[];
  const int wid  = threadIdx.x >> 5;
  const int lane = threadIdx.x & 31;
  const bool hi  = lane >= 16;
  const int o    = hi ? 2 : 0;
  const int nl   = lane & 15;
  const int wcol = blockIdx.x * 8 + wid;
  const int tileM = blockIdx.y * 16;
  const int S  = N * Kk;
  const int ST = S >> 4;

  stage_wband(sW, W, tileM, Cin);
  const float* Arow = sW + nl * Cin;      // LDS-resident A fragments

  float aS[8], aQ[8];
#pragma unroll
  for (int r = 0; r < 8; ++r) { aS[r] = 0.f; aQ[r] = 0.f; }

  for (int ct = wcol; ct < ST; ct += WTOT) {
    const int s = (ct << 4) + nl;
    const int n = s / Kk;
    const int j = (MODE == 0) ? s : idx[s];

    v8f a0 = {0.f,0.f,0.f,0.f,0.f,0.f,0.f,0.f};
    v8f a1 = a0, a2 = a0, u0 = a0, u1 = a0, u2 = a0;
    vn_mm<MODE, false>(X, Csrc, Nsrc, Cin, n, j, s, o, Arow, Arow, 0.f,
                       a0, a1, a2, u0, u1, u2);
#pragma unroll
    for (int r = 0; r < 8; ++r) {
      float nv = sqrtf(a0[r]*a0[r] + a1[r]*a1[r] + a2[r]*a2[r]) + EPSc;
      aS[r] += nv; aQ[r] += nv * nv;
    }
  }
  // reduce within each half-wave (lanes 0-15 -> ch tileM+r; 16-31 -> +8)
#pragma unroll
  for (int r = 0; r < 8; ++r) {
    float s1 = aS[r], s2 = aQ[r];
    for (int off = 1; off < 16; off <<= 1) {
      s1 += __shfl_xor(s1, off, 32);
      s2 += __shfl_xor(s2, off, 32);
    }
    if (nl == 0) {
      int ch = tileM + r + (hi ? 8 : 0);
      partial[(wcol * Cout + ch) * 2 + 0] = s1;
      partial[(wcol * Cout + ch) * 2 + 1] = s2;
    }
  }
}

__global__ void vn_reduce_kernel(const float* __restrict__ partial, int Cout,
                                 int WTOT, float invS,
                                 float* __restrict__ meanA, float* __restrict__ istdA) {
  int ch = blockIdx.x * blockDim.x + threadIdx.x;
  if (ch >= Cout) return;
  float s = 0.f, q = 0.f;
  for (int w = 0; w < WTOT; ++w) {
    s += partial[(w * Cout + ch) * 2 + 0];
    q += partial[(w * Cout + ch) * 2 + 1];
  }
  float m = s * invS;
  float v = q * invS - m * m;
  meanA[ch] = m;
  istdA[ch] = 1.0f / sqrtf(v + BN_EPSc);
}

// ---------------------------------------------------------------------------
// Pass 2: fused GEMM (p and d projections) + BN + VN-LeakyReLU epilogue.
// ---------------------------------------------------------------------------
template <int MODE>
__global__ void __launch_bounds__(256)
vn_finalize_kernel(const float* __restrict__ X, const int* __restrict__ idx,
                   int Csrc, int Nsrc, int Cin, int N, int Kk,
                   const float* __restrict__ W, int Cout,
                   const float* __restrict__ Wd, int Cd,
                   const float* __restrict__ gamma, const float* __restrict__ beta,
                   const float* __restrict__ meanA, const float* __restrict__ istdA,
                   float* __restrict__ OUT) {
  extern __shared__ float sW[];
  const int wid  = threadIdx.x >> 5;
  const int lane = threadIdx.x & 31;
  const bool hi  = lane >= 16;
  const int o    = hi ? 2 : 0;
  const int nl   = lane & 15;
  const int S  = N * Kk;
  const int ST = S >> 4;
  const int tileM = blockIdx.y * 16;

  stage_wband(sW, W, tileM, Cin);         // all threads participate (barrier)

  const int ct = blockIdx.x * 8 + wid;
  if (ct >= ST) return;
  const int mA = tileM + nl;
  const int mD = (Cd == 1) ? nl : mA;           // shared dir channel -> row 0
  const float dmask = (mD < Cd) ? 1.f : 0.f;    // branch-free Wd guard
  const float* Arow = sW + nl * Cin;            // LDS-resident A fragments
  const float* Drow = Wd + ((mD < Cd) ? mD : 0) * Cin;
  __builtin_prefetch(Drow, 0, 3);

  const int s = (ct << 4) + nl;
  const int n = s / Kk;
  const int j = (MODE == 0) ? s : idx[s];

  v8f p0 = {0.f,0.f,0.f,0.f,0.f,0.f,0.f,0.f};
  v8f p1 = p0, p2 = p0, d0 = p0, d1 = p0, d2 = p0;
  vn_mm<MODE, true>(X, Csrc, Nsrc, Cin, n, j, s, o, Arow, Drow, dmask,
                    p0, p1, p2, d0, d1, d2);

  float bda = 0.f, bdb = 0.f, bdc = 0.f;
  if (Cd == 1) {                 // broadcast channel-0 direction to all rows
    bda = __shfl(d0[0], nl, 32);
    bdb = __shfl(d1[0], nl, 32);
    bdc = __shfl(d2[0], nl, 32);
  }

#pragma unroll
  for (int r = 0; r < 8; ++r) {
    const int ch = tileM + r + (hi ? 8 : 0);
    float x0 = p0[r], x1 = p1[r], x2 = p2[r];
    float nrm = sqrtf(x0*x0 + x1*x1 + x2*x2) + EPSc;
    float nbn = gamma[ch] * (nrm - meanA[ch]) * istdA[ch] + beta[ch];
    float sc  = nbn / nrm;
    float q0 = x0 * sc, q1 = x1 * sc, q2 = x2 * sc;
    float e0 = (Cd == 1) ? bda : d0[r];
    float e1 = (Cd == 1) ? bdb : d1[r];
    float e2 = (Cd == 1) ? bdc : d2[r];
    float dot = q0*e0 + q1*e1 + q2*e2;
    float dsq = e0*e0 + e1*e1 + e2*e2;
    float inv = dot / (dsq + EPSc);
    float r0 = (dot >= 0.f) ? q0 : q0 - inv * e0;
    float r1 = (dot >= 0.f) ? q1 : q1 - inv * e1;
    float r2 = (dot >= 0.f) ? q2 : q2 - inv * e2;
    int base = ch * 3 * S;
    OUT[base + 0 * S + s] = NSc * q0 + (1.f - NSc) * r0;
    OUT[base + 1 * S + s] = NSc * q1 + (1.f - NSc) * r1;
    OUT[base + 2 * S + s] = NSc * q2 + (1.f - NSc) * r2;
  }
}

__global__ void vn_mean_kernel(const float* __restrict__ of, int C3, int N, int Kk,
                               float* __restrict__ dst) {
  int t = blockIdx.x * blockDim.x + threadIdx.x;
  if (t >= C3 * N) return;
  int ch = t / N;
  int n  = t - ch * N;
  const float* row = of + (size_t)ch * N * Kk + (size_t)n * Kk;
  float s = 0.f;
  for (int k = 0; k < Kk; ++k) s += row[k];
  dst[t] = s / (float)Kk;
}

// ---------------------------------------------------------------------------
// kNN in feature space: one wave per query, per-lane top-K lists merged via
// wave shfl-min.  X is (CD, N) feature-major (coalesced across lanes).
// ---------------------------------------------------------------------------
template <int KT>
__global__ void __launch_bounds__(256)
knn_feat_kernel(const float* __restrict__ X, int CD, int N, int* __restrict__ idx_out) {
  __shared__ float smem[12288];
  int* smemi = (int*)smem;
  const int wid  = threadIdx.x >> 5;
  const int lane = threadIdx.x & 31;
  const int q = blockIdx.x * 8 + wid;
  const bool qv = q < N;

  for (int c = lane; c < CD; c += 32)
    smem[wid * CD + c] = qv ? X[c * N + q] : 0.f;
  __syncthreads();

  float bd[KT]; int bi[KT];
#pragma unroll
  for (int t = 0; t < KT; ++t) { bd[t] = 1e30f; bi[t] = 0; }

  for (int jb = 0; jb < N; jb += 32) {
    int jj = jb + lane;
    float d = 0.f;
    for (int c = 0; c < CD; ++c) {
      float diff = X[c * N + jj] - smem[wid * CD + c];
      d += diff * diff;
    }
    if (d < bd[KT - 1]) {
      bd[KT - 1] = d; bi[KT - 1] = jj;
#pragma unroll
      for (int t = KT - 1; t > 0; --t) {
        if (bd[t] < bd[t - 1]) {
          float tv = bd[t]; bd[t] = bd[t - 1]; bd[t - 1] = tv;
          int ti = bi[t]; bi[t] = bi[t - 1]; bi[t - 1] = ti;
        }
      }
    }
  }
  __syncthreads();  // done with query vectors; reuse smem
#pragma unroll
  for (int t = 0; t < KT; ++t) {
    smem [(wid * 32 + lane) * KT + t]        = bd[t];
    smemi[5120 + (wid * 32 + lane) * KT + t] = bi[t];
  }
  __syncthreads();

  if (qv) {
    int cur = 0;
    for (int r = 0; r < KT; ++r) {
      float v  = (cur < KT) ? smem[(wid * 32 + lane) * KT + cur] : 1e30f;
      int   wl = lane;
      for (int off = 16; off; off >>= 1) {
        float ov = __shfl_xor(v, off, 32);
        int  owl = __shfl_xor(wl, off, 32);
        if (ov < v || (ov == v && owl < wl)) { v = ov; wl = owl; }
      }
      int hidx = (cur < KT) ? smemi[5120 + (wid * 32 + lane) * KT + cur] : 0;
      int widx = __shfl(hidx, wl, 32);
      if (lane == wl) cur++;
      if (lane == 0) idx_out[q * KT + r] = widx;
    }
  }
}

// kNN on xyz coordinates (row-major (n,3) clouds); optional interp weights.
template <int KT, bool WEIGHTS>
__global__ void knn_xyz_kernel(const float* __restrict__ qp, int M,
                               const float* __restrict__ pts, int Nn,
                               int* __restrict__ idx_out, float* __restrict__ w_out) {
  int i = blockIdx.x * blockDim.x + threadIdx.x;
  if (i >= M) return;
  float qx = qp[i * 3], qy = qp[i * 3 + 1], qz = qp[i * 3 + 2];
  float bd[KT]; int bi[KT];
#pragma unroll
  for (int t = 0; t < KT; ++t) { bd[t] = 1e30f; bi[t] = 0; }
  for (int jj = 0; jj < Nn; ++jj) {
    float dx = pts[jj * 3] - qx, dy = pts[jj * 3 + 1] - qy, dz = pts[jj * 3 + 2] - qz;
    float d = dx * dx + dy * dy + dz * dz;
    if (d < bd[KT - 1]) {
      bd[KT - 1] = d; bi[KT - 1] = jj;
#pragma unroll
      for (int t = KT - 1; t > 0; --t) {
        if (bd[t] < bd[t - 1]) {
          float tv = bd[t]; bd[t] = bd[t - 1]; bd[t - 1] = tv;
          int ti = bi[t]; bi[t] = bi[t - 1]; bi[t - 1] = ti;
        }
      }
    }
  }
  if (WEIGHTS) {
    float w[KT], ws = 0.f;
#pragma unroll
    for (int t = 0; t < KT; ++t) { w[t] = 1.0f / (bd[t] + 1e-8f); ws += w[t]; }
#pragma unroll
    for (int t = 0; t < KT; ++t) { w_out[i * KT + t] = w[t] / ws; idx_out[i * KT + t] = bi[t]; }
  } else {
#pragma unroll
    for (int t = 0; t < KT; ++t) idx_out[i * KT + t] = bi[t];
  }
}

// Farthest point sampling: single workgroup, first-index tie-break argmax.
__global__ void fps_kernel(const float* __restrict__ p, int n, int m,
                           int* __restrict__ out_idx, float* __restrict__ dists) {
  __shared__ float bv[256];
  __shared__ int   bix[256];
  __shared__ float sp[3];
  __shared__ int   slast;
  const int t = threadIdx.x;
  for (int i = t; i < n; i += 256) dists[i] = 1e10f;
  if (t == 0) { out_idx[0] = 0; slast = 0; }
  __syncthreads();
  for (int it = 1; it < m; ++it) {
    if (t == 0) { int l = slast; sp[0] = p[l*3]; sp[1] = p[l*3+1]; sp[2] = p[l*3+2]; }
    __syncthreads();
    float bestv = -1e30f; int besti = 0x7fffffff;
    for (int i = t; i < n; i += 256) {
      float dx = p[i*3] - sp[0], dy = p[i*3+1] - sp[1], dz = p[i*3+2] - sp[2];
      float d = dx*dx + dy*dy + dz*dz;
      float nd = fminf(dists[i], d);
      dists[i] = nd;
      if (nd > bestv) { bestv = nd; besti = i; }
    }
    bv[t] = bestv; bix[t] = besti;
    __syncthreads();
    for (int off = 128; off; off >>= 1) {
      if (t < off) {
        float ov = bv[t + off]; int oi = bix[t + off];
        if (ov > bv[t] || (ov == bv[t] && oi < bix[t])) { bv[t] = ov; bix[t] = oi; }
      }
      __syncthreads();
    }
    if (t == 0) { out_idx[it] = bix[0]; slast = bix[0]; }
    __syncthreads();
  }
}

__global__ void transpose3_kernel(const float* __restrict__ x, float* __restrict__ out, int N) {
  int i = blockIdx.x * blockDim.x + threadIdx.x;
  if (i < 3 * N) { int d = i / N; int n = i - d * N; out[i] = x[n * 3 + d]; }
}

__global__ void gather_pts_kernel(const float* __restrict__ p, const int* __restrict__ idx,
                                  int m, float* __restrict__ out) {
  int i = blockIdx.x * blockDim.x + threadIdx.x;
  if (i < m) {
    int jj = idx[i];
    out[i*3+0] = p[jj*3+0]; out[i*3+1] = p[jj*3+1]; out[i*3+2] = p[jj*3+2];
  }
}

__global__ void interp_apply_kernel(float* __restrict__ dst, const float* __restrict__ src,
                                    const int* __restrict__ idx3, const float* __restrict__ w3,
                                    int C3, int Nd, int Nsrc) {
  int t = blockIdx.x * blockDim.x + threadIdx.x;
  if (t >= C3 * Nd) return;
  int ch = t / Nd;
  int i  = t - ch * Nd;
  const float* row = src + (size_t)ch * Nsrc;
  float a = w3[i*3+0] * row[idx3[i*3+0]] +
            w3[i*3+1] * row[idx3[i*3+1]] +
            w3[i*3+2] * row[idx3[i*3+2]];
  dst[t] += a;
}

// ---------------------------------------------------------------------------
// Host orchestration
// ---------------------------------------------------------------------------
struct Lay { const float *W, *Wd, *g, *b; int Cout, Cd; };

struct Ws {
  float *outfull, *x1, *x2, *x3, *x4, *x5, *x6, *x7, *xtd, *bufu, *xin3;
  float *p2, *p3, *p4, *w3, *fpsd, *part, *statm, *stati;
  int *idxg, *idxq, *idxf, *idx3;
};

static void run_vn(const float* X, int Csrc, int Nsrc, int mode, const int* idx,
                   int N, int Kk, const Lay& L, float* dst, const Ws& w, hipStream_t s) {
  const int Cin = (mode == 1) ? 2 * Csrc : Csrc;
  const int S = N * Kk;
  const int ST = S / 16;
  const int WTOT = 512;
  const size_t shmem = (size_t)16 * Cin * sizeof(float);  // LDS weight band
  dim3 bs(256);
  dim3 gs(64, L.Cout / 16);
  dim3 gf((ST + 7) / 8, L.Cout / 16);
  float* out = (Kk == 1) ? dst : w.outfull;

  switch (mode) {
    case 0:
      vn_stats_kernel<0><<<gs, bs, shmem, s>>>(X, idx, Csrc, Nsrc, Cin, N, Kk, L.W, L.Cout, w.part, WTOT);
      break;
    case 1:
      vn_stats_kernel<1><<<gs, bs, shmem, s>>>(X, idx, Csrc, Nsrc, Cin, N, Kk, L.W, L.Cout, w.part, WTOT);
      break;
    default:
      vn_stats_kernel<2><<<gs, bs, shmem, s>>>(X, idx, Csrc, Nsrc, Cin, N, Kk, L.W, L.Cout, w.part, WTOT);
      break;
  }
  vn_reduce_kernel<<<dim3((L.Cout + 255) / 256), 256, 0, s>>>(w.part, L.Cout, WTOT,
                                                              1.0f / (float)S, w.statm, w.stati);
  switch (mode) {
    case 0:
      vn_finalize_kernel<0><<<gf, bs, shmem, s>>>(X, idx, Csrc, Nsrc, Cin, N, Kk, L.W, L.Cout,
                                                  L.Wd, L.Cd, L.g, L.b, w.statm, w.stati, out);
      break;
    case 1:
      vn_finalize_kernel<1><<<gf, bs, shmem, s>>>(X, idx, Csrc, Nsrc, Cin, N, Kk, L.W, L.Cout,
                                                  L.Wd, L.Cd, L.g, L.b, w.statm, w.stati, out);
      break;
    default:
      vn_finalize_kernel<2><<<gf, bs, shmem, s>>>(X, idx, Csrc, Nsrc, Cin, N, Kk, L.W, L.Cout,
                                                  L.Wd, L.Cd, L.g, L.b, w.statm, w.stati, out);
      break;
  }
  if (Kk > 1) {
    int tot = L.Cout * 3 * N;
    vn_mean_kernel<<<(tot + 255) / 256, 256, 0, s>>>(w.outfull, L.Cout * 3, N, Kk, dst);
  }
}

extern "C" void kernel_launch(void* const* d_in, const int* in_sizes, int n_in,
                              void* d_out, int out_size, void* d_ws, size_t ws_size,
                              hipStream_t stream) {
  (void)n_in; (void)out_size; (void)ws_size;

  // ---- input mapping: insertion order vs pytree-sorted order ----
  static const int coutTab[18] = {64,64,128,128,256,256,512,512,256,256,256,128,128,128,64,64,64,64};
  static const int sortedPos[18] = {0,9,1,10,2,11,3,4,12,13,5,14,15,6,16,17,7,8};
  const bool insertion = (in_sizes[0] == 8192 * 3);
  const float* xin;
  Lay L[18];
  if (insertion) {
    xin = (const float*)d_in[0];
    for (int i = 0; i < 18; ++i) {
      int base = 1 + 4 * i;
      L[i].W = (const float*)d_in[base + 0]; L[i].Wd = (const float*)d_in[base + 1];
      L[i].g = (const float*)d_in[base + 2]; L[i].b  = (const float*)d_in[base + 3];
    }
  } else {
    xin = (const float*)d_in[72];
    for (int i = 0; i < 18; ++i) {
      int base = 4 * sortedPos[i];
      L[i].W = (const float*)d_in[base + 0]; L[i].Wd = (const float*)d_in[base + 1];
      L[i].b = (const float*)d_in[base + 2]; L[i].g  = (const float*)d_in[base + 3];
    }
  }
  for (int i = 0; i < 18; ++i) { L[i].Cout = coutTab[i]; L[i].Cd = (i == 17) ? 1 : coutTab[i]; }
  enum {CONV1=0,DS1,CONV2,DS2,CONV3,DS3,CONV4,CONV5,UP1M1,UP1M2,CONV6,UP2M1,UP2M2,CONV7,UP3M1,UP3M2,CONV8,CONV9};

  // ---- workspace bump allocation (floats) ----
  float* base = (float*)d_ws;
  size_t o = 0;
  auto alloc = [&](size_t nf) { float* r = base + o; o += (nf + 63) & ~(size_t)63; return r; };
  Ws w;
  w.outfull = alloc(31457280);                 // max Cout*3*S = 125.8 MB
  w.x1 = alloc(1572864); w.x2 = alloc(1572864); w.x3 = alloc(1572864); w.x4 = alloc(1572864);
  w.x5 = alloc(1572864); w.x6 = alloc(1572864); w.x7 = alloc(1572864);
  w.xtd = alloc(786432); w.bufu = alloc(786432);
  w.xin3 = alloc(24576);
  w.p2 = alloc(12288); w.p3 = alloc(6144); w.p4 = alloc(3072);
  w.idxg = (int*)alloc(163840); w.idxq = (int*)alloc(65536); w.idxf = (int*)alloc(4096);
  w.idx3 = (int*)alloc(24576);  w.w3 = alloc(24576);
  w.fpsd = alloc(8192);
  w.part = alloc(524288);
  w.statm = alloc(512); w.stati = alloc(512);

  hipStream_t s = stream;
  auto g1 = [](int n) { return dim3((n + 255) / 256); };

  // (1,3,8192) feature-major copy of coords
  transpose3_kernel<<<g1(3 * 8192), 256, 0, s>>>(xin, w.xin3, 8192);

  // conv1: graph features on coords
  knn_feat_kernel<20><<<8192 / 8, 256, 0, s>>>(w.xin3, 3, 8192, w.idxg);
  run_vn(w.xin3, 1, 8192, 1, w.idxg, 8192, 20, L[CONV1], w.x1, w, s);

  // transition_down 1
  fps_kernel<<<1, 256, 0, s>>>(xin, 8192, 4096, w.idxf, w.fpsd);
  gather_pts_kernel<<<g1(4096), 256, 0, s>>>(xin, w.idxf, 4096, w.p2);
  knn_xyz_kernel<16, false><<<g1(4096), 256, 0, s>>>(w.p2, 4096, xin, 8192, w.idxq, nullptr);
  run_vn(w.x1, 64, 8192, 2, w.idxq, 4096, 16, L[DS1], w.xtd, w, s);
  knn_feat_kernel<20><<<4096 / 8, 256, 0, s>>>(w.xtd, 192, 4096, w.idxg);
  run_vn(w.xtd, 64, 4096, 1, w.idxg, 4096, 20, L[CONV2], w.x2, w, s);

  // transition_down 2
  fps_kernel<<<1, 256, 0, s>>>(w.p2, 4096, 2048, w.idxf, w.fpsd);
  gather_pts_kernel<<<g1(2048), 256, 0, s>>>(w.p2, w.idxf, 2048, w.p3);
  knn_xyz_kernel<16, false><<<g1(2048), 256, 0, s>>>(w.p3, 2048, w.p2, 4096, w.idxq, nullptr);
  run_vn(w.x2, 128, 4096, 2, w.idxq, 2048, 16, L[DS2], w.xtd, w, s);
  knn_feat_kernel<20><<<2048 / 8, 256, 0, s>>>(w.xtd, 384, 2048, w.idxg);
  run_vn(w.xtd, 128, 2048, 1, w.idxg, 2048, 20, L[CONV3], w.x3, w, s);

  // transition_down 3
  fps_kernel<<<1, 256, 0, s>>>(w.p3, 2048, 1024, w.idxf, w.fpsd);
  gather_pts_kernel<<<g1(1024), 256, 0, s>>>(w.p3, w.idxf, 1024, w.p4);
  knn_xyz_kernel<16, false><<<g1(1024), 256, 0, s>>>(w.p4, 1024, w.p3, 2048, w.idxq, nullptr);
  run_vn(w.x3, 256, 2048, 2, w.idxq, 1024, 16, L[DS3], w.xtd, w, s);
  knn_feat_kernel<20><<<1024 / 8, 256, 0, s>>>(w.xtd, 768, 1024, w.idxg);
  run_vn(w.xtd, 256, 1024, 1, w.idxg, 1024, 20, L[CONV4], w.x4, w, s);

  // conv5 (bottleneck)
  knn_feat_kernel<20><<<1024 / 8, 256, 0, s>>>(w.x4, 1536, 1024, w.idxg);
  run_vn(w.x4, 512, 1024, 1, w.idxg, 1024, 20, L[CONV5], w.x4, w, s);

  // up 1
  run_vn(w.x3, 256, 2048, 0, nullptr, 2048, 1, L[UP1M1], w.x5, w, s);
  run_vn(w.x4, 512, 1024, 0, nullptr, 1024, 1, L[UP1M2], w.bufu, w, s);
  knn_xyz_kernel<3, true><<<g1(2048), 256, 0, s>>>(w.p3, 2048, w.p4, 1024, w.idx3, w.w3);
  interp_apply_kernel<<<g1(256 * 3 * 2048), 256, 0, s>>>(w.x5, w.bufu, w.idx3, w.w3, 768, 2048, 1024);
  knn_feat_kernel<20><<<2048 / 8, 256, 0, s>>>(w.x5, 768, 2048, w.idxg);
  run_vn(w.x5, 256, 2048, 1, w.idxg, 2048, 20, L[CONV6], w.x5, w, s);

  // up 2
  run_vn(w.x2, 128, 4096, 0, nullptr, 4096, 1, L[UP2M1], w.x6, w, s);
  run_vn(w.x5, 256, 2048, 0, nullptr, 2048, 1, L[UP2M2], w.bufu, w, s);
  knn_xyz_kernel<3, true><<<g1(4096), 256, 0, s>>>(w.p2, 4096, w.p3, 2048, w.idx3, w.w3);
  interp_apply_kernel<<<g1(128 * 3 * 4096), 256, 0, s>>>(w.x6, w.bufu, w.idx3, w.w3, 384, 4096, 2048);
  knn_feat_kernel<20><<<4096 / 8, 256, 0, s>>>(w.x6, 384, 4096, w.idxg);
  run_vn(w.x6, 128, 4096, 1, w.idxg, 4096, 20, L[CONV7], w.x6, w, s);

  // up 3
  run_vn(w.x1, 64, 8192, 0, nullptr, 8192, 1, L[UP3M1], w.x7, w, s);
  run_vn(w.x6, 128, 4096, 0, nullptr, 4096, 1, L[UP3M2], w.bufu, w, s);
  knn_xyz_kernel<3, true><<<g1(8192), 256, 0, s>>>(xin, 8192, w.p2, 4096, w.idx3, w.w3);
  interp_apply_kernel<<<g1(64 * 3 * 8192), 256, 0, s>>>(w.x7, w.bufu, w.idx3, w.w3, 192, 8192, 4096);
  knn_feat_kernel<20><<<8192 / 8, 256, 0, s>>>(w.x7, 192, 8192, w.idxg);
  run_vn(w.x7, 64, 8192, 1, w.idxg, 8192, 20, L[CONV8], w.x7, w, s);

  // conv9 -> output (64,3,8192) f32, written directly by finalize (K=1)
  run_vn(w.x7, 64, 8192, 0, nullptr, 8192, 1, L[CONV9], (float*)d_out, w, s);
}